// GCNRegressor_2327872274533
// MI455X (gfx1250) — compile-verified
//
#include <hip/hip_runtime.h>

// GCN regressor for MI455X (gfx1250, wave32).
// F_IN = HID = 128 hardcoded (matches reference).

typedef float v2f __attribute__((ext_vector_type(2)));
typedef float v8f __attribute__((ext_vector_type(8)));

#define HID 128

// ---------------- degree / normalization ----------------

__global__ void deg_init_kernel(float* __restrict__ deg, int n) {
    int i = blockIdx.x * blockDim.x + threadIdx.x;
    if (i < n) deg[i] = 1.0f;  // self-loop contributes 1 to every node
}

__global__ void deg_edge_kernel(const int* __restrict__ dst, float* __restrict__ deg, int e) {
    int i = blockIdx.x * blockDim.x + threadIdx.x;
    if (i < e) atomicAdd(&deg[dst[i]], 1.0f);
}

__global__ void dis_kernel(const float* __restrict__ deg, float* __restrict__ dis, int n) {
    int i = blockIdx.x * blockDim.x + threadIdx.x;
    if (i < n) dis[i] = rsqrtf(deg[i]);  // deg >= 1 always (self loops)
}

// ---------------- dense transform: Y = act(X) @ W via WMMA f32 16x16x4 ----------------
// 4 waves per block; each wave owns TWO 16-col output tiles (cols w*16 and w*16+64),
// so one A fragment feeds two WMMAs (A reuse, 2 independent accumulator chains).
// Grid.x = N/16 row strips (N = 50000 = 3125 * 16, exact).

template<bool RELU>
__global__ __launch_bounds__(128) void gemm_wmma_f32(
    const float* __restrict__ X, const float* __restrict__ W,
    float* __restrict__ Y)
{
    const int wave = threadIdx.x >> 5;    // 0..3
    const int lane = threadIdx.x & 31;
    const int half = lane >> 4;           // 0: K pair {0,1}; 1: K pair {2,3}
    const int l15  = lane & 15;
    const int row0 = blockIdx.x * 16;
    const int colA = wave * 16;           // first column tile
    const int colB = colA + 64;           // second column tile
    const int arow = row0 + l15;

    v8f c0 = {0.f, 0.f, 0.f, 0.f, 0.f, 0.f, 0.f, 0.f};
    v8f c1 = {0.f, 0.f, 0.f, 0.f, 0.f, 0.f, 0.f, 0.f};

    #pragma unroll 4
    for (int k0 = 0; k0 < HID; k0 += 4) {
        const int ka = k0 + half * 2;
        // A fragment (16x4, row M=l15): this lane holds X[arow][ka], X[arow][ka+1]
        float a0 = X[arow * HID + ka];
        float a1 = X[arow * HID + ka + 1];
        if (RELU) { a0 = fmaxf(a0, 0.f); a1 = fmaxf(a1, 0.f); }
        v2f a = { a0, a1 };
        // B fragments (4x16): row K striped across lanes within a VGPR
        v2f b0 = { W[ka * HID + colA + l15], W[(ka + 1) * HID + colA + l15] };
        v2f b1 = { W[ka * HID + colB + l15], W[(ka + 1) * HID + colB + l15] };
        // (neg_a, A, neg_b, B, c_mod, C, reuse_a, reuse_b)
        c0 = __builtin_amdgcn_wmma_f32_16x16x4_f32(
                 false, a, false, b0, (short)0, c0, false, false);
        c1 = __builtin_amdgcn_wmma_f32_16x16x4_f32(
                 false, a, false, b1, (short)0, c1, false, false);
    }

    // D layout: VGPR i -> M = i + 8*half, N = l15
    #pragma unroll
    for (int i = 0; i < 8; ++i) {
        const int r = (row0 + half * 8 + i) * HID;
        Y[r + colA + l15] = c0[i];
        Y[r + colB + l15] = c1[i];
    }
}

// ---------------- aggregation ----------------
// out[i][:] = h[i][:] * dis[i]^2 + bias   (self-loop term + bias, also zero-inits)
__global__ void agg_init_kernel(const float* __restrict__ H, const float* __restrict__ dis,
                                const float* __restrict__ bias, float* __restrict__ OUT, int n)
{
    int idx = blockIdx.x * blockDim.x + threadIdx.x;   // n * 32 threads, float4 each
    if (idx >= n * 32) return;
    int node = idx >> 5, c4 = (idx & 31) * 4;
    float s = dis[node]; float ss = s * s;
    float4 h = *(const float4*)(H + node * HID + c4);
    float4 b = *(const float4*)(bias + c4);
    float4 o = { h.x * ss + b.x, h.y * ss + b.y, h.z * ss + b.z, h.w * ss + b.w };
    *(float4*)(OUT + node * HID + c4) = o;
}

// out[dst] += h[src] * dis[src] * dis[dst]   — one wave per edge, float4 per lane
__global__ __launch_bounds__(256) void edge_scatter_kernel(
    const int* __restrict__ src, const int* __restrict__ dst,
    const float* __restrict__ dis, const float* __restrict__ H,
    float* __restrict__ OUT, int e_cnt)
{
    int e = blockIdx.x * 8 + (threadIdx.x >> 5);
    if (e >= e_cnt) return;
    int lane = threadIdx.x & 31;
    int s = src[e], d = dst[e];
    float nrm = dis[s] * dis[d];
    const float4 h = *(const float4*)(H + s * HID + lane * 4);
    float* o = OUT + d * HID + lane * 4;
    atomicAdd(o + 0, h.x * nrm);
    atomicAdd(o + 1, h.y * nrm);
    atomicAdd(o + 2, h.z * nrm);
    atomicAdd(o + 3, h.w * nrm);
}

// ---------------- pooling + final FC ----------------

__global__ void pool_zero_kernel(float* __restrict__ sums, float* __restrict__ cnts, int g) {
    int i = blockIdx.x * blockDim.x + threadIdx.x;
    if (i < g * HID) sums[i] = 0.f;
    if (i < g) cnts[i] = 0.f;
}

__global__ void pool_kernel(const float* __restrict__ H, const int* __restrict__ batch,
                            float* __restrict__ sums, float* __restrict__ cnts, int n)
{
    int node = blockIdx.x * 8 + (threadIdx.x >> 5);
    if (node >= n) return;
    int lane = threadIdx.x & 31;
    int g = batch[node];
    float4 h = *(const float4*)(H + node * HID + lane * 4);
    float* sp = sums + g * HID + lane * 4;
    atomicAdd(sp + 0, h.x);
    atomicAdd(sp + 1, h.y);
    atomicAdd(sp + 2, h.z);
    atomicAdd(sp + 3, h.w);
    if (lane == 0) atomicAdd(&cnts[g], 1.0f);
}

__global__ void final_kernel(const float* __restrict__ sums, const float* __restrict__ cnts,
                             const float* __restrict__ fcW, const float* __restrict__ fcb,
                             float* __restrict__ out, int g_cnt)
{
    int g = blockIdx.x * blockDim.x + threadIdx.x;
    if (g >= g_cnt) return;
    float acc = 0.f;
    #pragma unroll 8
    for (int k = 0; k < HID; ++k) acc += sums[g * HID + k] * fcW[k];
    out[g] = acc / fmaxf(cnts[g], 1.0f) + fcb[0];
}

// ---------------- host launcher ----------------

extern "C" void kernel_launch(void* const* d_in, const int* in_sizes, int n_in,
                              void* d_out, int out_size, void* d_ws, size_t ws_size,
                              hipStream_t stream) {
    const float* x     = (const float*)d_in[0];
    const int*   ei    = (const int*)  d_in[1];
    const int*   batch = (const int*)  d_in[2];
    const float* W0    = (const float*)d_in[3];
    const float* b0    = (const float*)d_in[4];
    const float* W1    = (const float*)d_in[5];
    const float* b1    = (const float*)d_in[6];
    const float* W2    = (const float*)d_in[7];
    const float* b2    = (const float*)d_in[8];
    const float* fcW   = (const float*)d_in[9];
    const float* fcb   = (const float*)d_in[10];
    float* out = (float*)d_out;

    const int N = in_sizes[0] / HID;   // 50000
    const int E = in_sizes[1] / 2;     // 800000
    const int G = out_size;            // 64
    const int* src = ei;
    const int* dst = ei + E;

    // carve workspace
    size_t off = 0;
    auto carve = [&](size_t elems) -> float* {
        float* p = (float*)((char*)d_ws + off);
        off += ((elems * sizeof(float)) + 255) & ~(size_t)255;
        return p;
    };
    float* deg  = carve(N);
    float* dis  = carve(N);
    float* hA   = carve((size_t)N * HID);  // gemm output (per layer)
    float* hB   = carve((size_t)N * HID);  // layer 0 & 2 aggregation output
    float* hC   = carve((size_t)N * HID);  // layer 1 aggregation output
    float* sums = carve((size_t)G * HID);
    float* cnts = carve(G);
    (void)ws_size; (void)n_in;

    const int TB = 256;
    const int nBlk   = (N + TB - 1) / TB;
    const int eBlk   = (E + TB - 1) / TB;
    const int rowBlk = N / 16;                 // 3125, exact
    const int ewBlk  = (E + 7) / 8;            // wave-per-edge
    const int nwBlk  = (N + 7) / 8;            // wave-per-node
    const int n32Blk = (N * 32 + TB - 1) / TB;

    // normalization: deg = in-degree + 1 (self loop); dis = deg^-1/2
    deg_init_kernel<<<nBlk, TB, 0, stream>>>(deg, N);
    deg_edge_kernel<<<eBlk, TB, 0, stream>>>(dst, deg, E);
    dis_kernel<<<nBlk, TB, 0, stream>>>(deg, dis, N);

    // layer 0: in = x (no relu)
    gemm_wmma_f32<false><<<rowBlk, 128, 0, stream>>>(x, W0, hA);
    agg_init_kernel<<<n32Blk, TB, 0, stream>>>(hA, dis, b0, hB, N);
    edge_scatter_kernel<<<ewBlk, TB, 0, stream>>>(src, dst, dis, hA, hB, E);

    // layer 1: in = relu(hB)
    gemm_wmma_f32<true><<<rowBlk, 128, 0, stream>>>(hB, W1, hA);
    agg_init_kernel<<<n32Blk, TB, 0, stream>>>(hA, dis, b1, hC, N);
    edge_scatter_kernel<<<ewBlk, TB, 0, stream>>>(src, dst, dis, hA, hC, E);

    // layer 2: in = relu(hC), no relu on output
    gemm_wmma_f32<true><<<rowBlk, 128, 0, stream>>>(hC, W2, hA);
    agg_init_kernel<<<n32Blk, TB, 0, stream>>>(hA, dis, b2, hB, N);
    edge_scatter_kernel<<<ewBlk, TB, 0, stream>>>(src, dst, dis, hA, hB, E);

    // global mean pool + FC
    pool_zero_kernel<<<(G * HID + TB - 1) / TB, TB, 0, stream>>>(sums, cnts, G);
    pool_kernel<<<nwBlk, TB, 0, stream>>>(hB, batch, sums, cnts, N);
    final_kernel<<<1, 64, 0, stream>>>(sums, cnts, fcW, fcb, out, G);
}